// GCN_33036888441460
// MI455X (gfx1250) — compile-verified
//
#include <hip/hip_runtime.h>
#include <math.h>

// ---------------------------------------------------------------------------
// GCN: 2x ChebConv(K=2) + ReLU, then linear head.
//   deg  = segment_sum(1, src);  dis = deg>0 ? deg^-0.5 : 0
//   w_e  = -(dis[src]*dis[dst])
//   Tx1[dst] += w_e * x[src]   (scatter-add, f32 atomics -> L2 resident)
//   h = relu(x@W[0] + Tx1@W[1] + b)
// Head GEMM (100000x16 @ 16x64) uses V_WMMA_F32_16X16X4_F32 (exact f32).
// ---------------------------------------------------------------------------

typedef __attribute__((ext_vector_type(2))) float v2f;
typedef __attribute__((ext_vector_type(8))) float v8f;

#define N_NODES 100000

__global__ void zero_kernel(float* __restrict__ p, int n) {
  int i = blockIdx.x * blockDim.x + threadIdx.x;
  if (i < n) p[i] = 0.0f;
}

__global__ void degree_kernel(const int* __restrict__ src,
                              float* __restrict__ deg, int E) {
  int e = blockIdx.x * blockDim.x + threadIdx.x;
  if (e < E) atomicAdd(deg + src[e], 1.0f);
}

__global__ void dis_kernel(const float* __restrict__ deg,
                           float* __restrict__ dis, int n) {
  int i = blockIdx.x * blockDim.x + threadIdx.x;
  if (i < n) {
    float d = deg[i];
    dis[i] = (d > 0.0f) ? (1.0f / sqrtf(d)) : 0.0f;
  }
}

// Layer-1 propagate: 8 features per edge. x[src] fetched as 2x float4,
// scatter-add into Tx1[dst*8..] (accumulator is 3.2 MB -> L2 resident).
__global__ void prop1_kernel(const int* __restrict__ src, const int* __restrict__ dst,
                             const float* __restrict__ dis, const float* __restrict__ x,
                             float* __restrict__ tx, int E) {
  int e = blockIdx.x * blockDim.x + threadIdx.x;
  if (e >= E) return;
  int s = src[e], d = dst[e];
  float w = -(dis[s] * dis[d]);
  const float4* xs = reinterpret_cast<const float4*>(x + (size_t)s * 8);
  float4 a = xs[0];
  float4 b = xs[1];
  float* t = tx + (size_t)d * 8;
  atomicAdd(t + 0, w * a.x); atomicAdd(t + 1, w * a.y);
  atomicAdd(t + 2, w * a.z); atomicAdd(t + 3, w * a.w);
  atomicAdd(t + 4, w * b.x); atomicAdd(t + 5, w * b.y);
  atomicAdd(t + 6, w * b.z); atomicAdd(t + 7, w * b.w);
}

// h1 = relu(x @ W1[0] + Tx1 @ W1[1] + b1)   (8 -> 3)
__global__ void layer1_kernel(const float* __restrict__ x, const float* __restrict__ tx,
                              const float* __restrict__ W1, const float* __restrict__ b1,
                              float* __restrict__ h1, int n) {
  int i = blockIdx.x * blockDim.x + threadIdx.x;
  if (i >= n) return;
  const float4* xp = reinterpret_cast<const float4*>(x + (size_t)i * 8);
  const float4* tp = reinterpret_cast<const float4*>(tx + (size_t)i * 8);
  float4 x0 = xp[0], x1 = xp[1], t0 = tp[0], t1 = tp[1];
  float xi[8] = {x0.x, x0.y, x0.z, x0.w, x1.x, x1.y, x1.z, x1.w};
  float ti[8] = {t0.x, t0.y, t0.z, t0.w, t1.x, t1.y, t1.z, t1.w};
#pragma unroll
  for (int h = 0; h < 3; ++h) {
    float acc = b1[h];
#pragma unroll
    for (int f = 0; f < 8; ++f)
      acc += xi[f] * W1[f * 3 + h] + ti[f] * W1[24 + f * 3 + h];
    h1[(size_t)i * 3 + h] = fmaxf(acc, 0.0f);
  }
}

// Layer-2 propagate: 3 features per edge.
__global__ void prop2_kernel(const int* __restrict__ src, const int* __restrict__ dst,
                             const float* __restrict__ dis, const float* __restrict__ h1,
                             float* __restrict__ tx, int E) {
  int e = blockIdx.x * blockDim.x + threadIdx.x;
  if (e >= E) return;
  int s = src[e], d = dst[e];
  float w = -(dis[s] * dis[d]);
  const float* hs = h1 + (size_t)s * 3;
  float* t = tx + (size_t)d * 3;
  atomicAdd(t + 0, w * hs[0]);
  atomicAdd(t + 1, w * hs[1]);
  atomicAdd(t + 2, w * hs[2]);
}

// h2 = relu(h1 @ W2[0] + Tx1b @ W2[1] + b2)   (3 -> 16)
__global__ void layer2_kernel(const float* __restrict__ h1, const float* __restrict__ tx,
                              const float* __restrict__ W2, const float* __restrict__ b2,
                              float* __restrict__ h2, int n) {
  int i = blockIdx.x * blockDim.x + threadIdx.x;
  if (i >= n) return;
  float hv[3], tv[3];
#pragma unroll
  for (int f = 0; f < 3; ++f) {
    hv[f] = h1[(size_t)i * 3 + f];
    tv[f] = tx[(size_t)i * 3 + f];
  }
#pragma unroll
  for (int h = 0; h < 16; ++h) {
    float acc = b2[h];
#pragma unroll
    for (int f = 0; f < 3; ++f)
      acc += hv[f] * W2[f * 16 + h] + tv[f] * W2[48 + f * 16 + h];
    h2[(size_t)i * 16 + h] = fmaxf(acc, 0.0f);
  }
}

// out = h2 @ Wl + bl using V_WMMA_F32_16X16X4_F32.
// One wave computes a 16-row x 64-col output tile:
//   4 N-tiles (j) x 4 K-steps (kk) = 16 WMMA ops, K=16 total.
// Fragment layouts per CDNA5 ISA (32-bit A 16x4):
//   A: lane l, half = l>=16: VGPR0 = A[l&15][kk + 2*half], VGPR1 = +1
//   B: lane l: VGPR0 = B[kk + 2*half][col], VGPR1 = B[kk + 2*half + 1][col]
//   C/D: VGPR r, lane l -> (M = r + 8*half, N = l&15)
__global__ void head_gemm_wmma(const float* __restrict__ h2, const float* __restrict__ Wl,
                               const float* __restrict__ bl, float* __restrict__ out,
                               int ntiles) {
  int gwave = (blockIdx.x * blockDim.x + threadIdx.x) >> 5;
  int lane  = threadIdx.x & 31;
  if (gwave >= ntiles) return;           // whole-wave uniform exit: EXEC stays all-1s
  int half = lane >> 4;
  int l16  = lane & 15;
  int row  = gwave * 16 + l16;

  v8f acc0 = {}, acc1 = {}, acc2 = {}, acc3 = {};
#pragma unroll
  for (int kk = 0; kk < 16; kk += 4) {
    int ka = kk + 2 * half;
    v2f a;
    a.x = h2[(size_t)row * 16 + ka];
    a.y = h2[(size_t)row * 16 + ka + 1];

    v2f b0, b1, b2, b3;
    b0.x = Wl[ka * 64 +  0 + l16]; b0.y = Wl[(ka + 1) * 64 +  0 + l16];
    b1.x = Wl[ka * 64 + 16 + l16]; b1.y = Wl[(ka + 1) * 64 + 16 + l16];
    b2.x = Wl[ka * 64 + 32 + l16]; b2.y = Wl[(ka + 1) * 64 + 32 + l16];
    b3.x = Wl[ka * 64 + 48 + l16]; b3.y = Wl[(ka + 1) * 64 + 48 + l16];

    acc0 = __builtin_amdgcn_wmma_f32_16x16x4_f32(false, a, false, b0, (short)0, acc0, false, false);
    acc1 = __builtin_amdgcn_wmma_f32_16x16x4_f32(false, a, false, b1, (short)0, acc1, false, false);
    acc2 = __builtin_amdgcn_wmma_f32_16x16x4_f32(false, a, false, b2, (short)0, acc2, false, false);
    acc3 = __builtin_amdgcn_wmma_f32_16x16x4_f32(false, a, false, b3, (short)0, acc3, false, false);
  }

  v8f accs[4] = {acc0, acc1, acc2, acc3};
#pragma unroll
  for (int j = 0; j < 4; ++j) {
    int col = j * 16 + l16;
    float bias = bl[col];
#pragma unroll
    for (int r = 0; r < 8; ++r) {
      int m = gwave * 16 + r + 8 * half;
      out[(size_t)m * 64 + col] = accs[j][r] + bias;
    }
  }
}

extern "C" void kernel_launch(void* const* d_in, const int* in_sizes, int n_in,
                              void* d_out, int out_size, void* d_ws, size_t ws_size,
                              hipStream_t stream) {
  const float* x  = (const float*)d_in[0];   // [N,8]
  const int*   ei = (const int*)d_in[1];     // [2,E] (src then dst)
  const float* W1 = (const float*)d_in[2];   // [2,8,3]
  const float* b1 = (const float*)d_in[3];   // [3]
  const float* W2 = (const float*)d_in[4];   // [2,3,16]
  const float* b2 = (const float*)d_in[5];   // [16]
  const float* Wl = (const float*)d_in[6];   // [16,64]
  const float* bl = (const float*)d_in[7];   // [64]
  float* out = (float*)d_out;                // [N,64]

  const int N = in_sizes[0] / 8;             // 100000
  const int E = in_sizes[1] / 2;             // 6400000
  const int* src = ei;
  const int* dst = ei + E;

  // Workspace layout (floats): deg[N] dis[N] tx1[8N] h1[3N] tx2[3N] h2[16N]
  float* ws  = (float*)d_ws;
  float* deg = ws;
  float* dis = ws + (size_t)N;
  float* tx1 = ws + (size_t)2  * N;
  float* h1  = ws + (size_t)10 * N;
  float* tx2 = ws + (size_t)13 * N;
  float* h2  = ws + (size_t)16 * N;

  const int B = 256;
  // Zero deg/dis/tx1/h1/tx2 (dis & h1 fully overwritten anyway; harmless).
  zero_kernel<<<(16 * N + B - 1) / B, B, 0, stream>>>(ws, 16 * N);

  degree_kernel<<<(E + B - 1) / B, B, 0, stream>>>(src, deg, E);
  dis_kernel<<<(N + B - 1) / B, B, 0, stream>>>(deg, dis, N);

  prop1_kernel<<<(E + B - 1) / B, B, 0, stream>>>(src, dst, dis, x, tx1, E);
  layer1_kernel<<<(N + B - 1) / B, B, 0, stream>>>(x, tx1, W1, b1, h1, N);

  prop2_kernel<<<(E + B - 1) / B, B, 0, stream>>>(src, dst, dis, h1, tx2, E);
  layer2_kernel<<<(N + B - 1) / B, B, 0, stream>>>(h1, tx2, W2, b2, h2, N);

  const int ntiles = N / 16;                 // 6250 (N is a multiple of 16)
  head_gemm_wmma<<<(ntiles * 32 + B - 1) / B, B, 0, stream>>>(h2, Wl, bl, out, ntiles);
}